// CollapseAwareEmbedding_63118839382516
// MI455X (gfx1250) — compile-verified
//
#include <hip/hip_runtime.h>
#include <hip/hip_bf16.h>
#include <cstddef>

typedef float v2f __attribute__((ext_vector_type(2)));
typedef float v4f __attribute__((ext_vector_type(4)));
typedef float v8f __attribute__((ext_vector_type(8)));

// Global row space: row 0 = collapse token.
// reg: 0 collapse, 1 hd[1,401), 2 mhc[401,701), 3 pep[701,716),
//      4 lv[716,826), 5 lj[826,876), 6 hv[876,976), 7 hj[976,1024)
__device__ __forceinline__ int region_of(int r) {
  return (r >= 1) + (r >= 401) + (r >= 701) + (r >= 716) +
         (r >= 826) + (r >= 876) + (r >= 976);
}

// ---------------------------------------------------------------------------
// Prep kernels (tiny)
// ---------------------------------------------------------------------------

// Stage rw weights into ws: rwTab[reg*2 + {0,1}], reg 1..7
__global__ void prep_rw(const float* a, const float* b, const float* c,
                        const float* d, const float* e, const float* f,
                        const float* g, float* __restrict__ rwTab) {
  int t = threadIdx.x;
  if (t < 16) {
    const float* ptrs[7] = {a, b, c, d, e, f, g};
    float v = 0.0f;
    if (t >= 2) v = ptrs[(t >> 1) - 1][t & 1];
    rwTab[t] = v;
  }
}

// Fused weight matrix W (96 x 256): rows 0-21 = W_seq, 22-85 = W_pos, 86-95 = 0
__global__ void build_w(const float* __restrict__ W_seq,
                        const float* __restrict__ W_pos,
                        float* __restrict__ W) {
  int row = blockIdx.x, col = threadIdx.x;
  float v = 0.0f;
  if (row < 22)      v = W_seq[row * 256 + col];
  else if (row < 86) v = W_pos[(row - 22) * 256 + col];
  W[row * 256 + col] = v;
}

// zvec table (32 x 128): zvec[pid] = concat(W_p1[pid/4]+b_p1, W_p2[pid%4]+b_p2)
__global__ void build_zvec(const float* __restrict__ W_p1,
                           const float* __restrict__ b_p1,
                           const float* __restrict__ W_p2,
                           const float* __restrict__ b_p2,
                           float* __restrict__ zvec) {
  int pid = blockIdx.x, c = threadIdx.x;
  float v = (c < 64) ? (W_p1[(pid >> 2) * 64 + c] + b_p1[c])
                     : (W_p2[(pid & 3) * 64 + (c - 64)] + b_p2[c - 64]);
  zvec[pid * 128 + c] = v;
}

// Fused input matrix X (1024 x 96):
//   cols 0-21  = rw[0] * aa (hd rows masked; row 0 all-zero)
//   cols 22-85 = rw[1] * pos_encoding(idx)
//   cols 86-95 = 0 (K padding to 96 = 24*4)
__global__ void build_x(
    const float* hd, const int* hd_idx, const float* mhc, const int* mhc_idx,
    const float* pep, const int* pep_idx, const float* lv, const int* lv_idx,
    const float* lj, const int* lj_idx, const float* hv, const int* hv_idx,
    const float* hj, const int* hj_idx, const int* __restrict__ mask,
    const float* rw_hd, const float* rw_mhc, const float* rw_pep,
    const float* rw_lv, const float* rw_lj, const float* rw_hv,
    const float* rw_hj, float* __restrict__ X) {
  int r = blockIdx.x;
  int c = threadIdx.x; // 0..95
  float v = 0.0f;
  if (r > 0) {
    int reg = region_of(r);
    const int starts[8] = {0, 1, 401, 701, 716, 826, 876, 976};
    const float* aa; const int* idx; const float* rw;
    switch (reg) {
      case 1:  aa = hd;  idx = hd_idx;  rw = rw_hd;  break;
      case 2:  aa = mhc; idx = mhc_idx; rw = rw_mhc; break;
      case 3:  aa = pep; idx = pep_idx; rw = rw_pep; break;
      case 4:  aa = lv;  idx = lv_idx;  rw = rw_lv;  break;
      case 5:  aa = lj;  idx = lj_idx;  rw = rw_lj;  break;
      case 6:  aa = hv;  idx = hv_idx;  rw = rw_hv;  break;
      default: aa = hj;  idx = hj_idx;  rw = rw_hj;  break;
    }
    int lr = r - starts[reg];
    if (c < 22) {
      float x = aa[lr * 22 + c];
      if (reg == 1 && mask[lr] == 1) x = 0.0f;
      v = rw[0] * x;
    } else if (c < 86) {
      int i = c - 22;
      float pos = (float)idx[lr];
      // 10000^(-i/64) = exp(-i/64 * ln(10000))
      float ang = pos * expf(-((float)i) * (9.210340371976184f / 64.0f));
      v = rw[1] * (((i & 1) == 0) ? sinf(ang) : cosf(ang));
    }
  }
  X[r * 96 + c] = v;
}

// ---------------------------------------------------------------------------
// s_out GEMM: (1024 x 96) @ (96 x 256) via V_WMMA_F32_16X16X4_F32
// One wave (32 threads) per 16x16 output tile; 24 WMMA steps of K=4.
// ---------------------------------------------------------------------------
__global__ void __launch_bounds__(32)
gemm_s(const float* __restrict__ X, const float* __restrict__ W,
       const float* __restrict__ b_seq, const float* __restrict__ b_pos,
       const float* __restrict__ token, const float* __restrict__ cw,
       const float* __restrict__ rwTab, float* __restrict__ out) {
  int lane = threadIdx.x;
  int l15  = lane & 15;
  int half = lane >> 4;                       // 0: K 0,1 | 1: K 2,3
  int rowBase = blockIdx.y * 16;
  int colBase = blockIdx.x * 16;

  // A 16x4 f32: lanes 0-15 rows M=0..15 K={0,1}; lanes 16-31 K={2,3}
  const float* Xp = X + (size_t)(rowBase + l15) * 96 + half * 2;
  // B 4x16 f32 (mirrored layout): N = lane&15, VGPR0/1 hold K rows {0,2}/{1,3}
  const float* Wp = W + (size_t)(half * 2) * 256 + colBase + l15;

  v8f acc = {};
#pragma unroll
  for (int k = 0; k < 24; ++k) {
    v2f a, b;
    a.x = Xp[0];
    a.y = Xp[1];
    b.x = Wp[0];
    b.y = Wp[256];
    acc = __builtin_amdgcn_wmma_f32_16x16x4_f32(
        /*neg_a=*/false, a, /*neg_b=*/false, b,
        /*c_mod=*/(short)0, acc, /*reuse_a=*/false, /*reuse_b=*/false);
    Xp += 4;
    Wp += 4 * 256;
  }

  // C/D layout: VGPR j -> M = j + 8*half, N = lane&15
  int col = colBase + l15;
#pragma unroll
  for (int vj = 0; vj < 8; ++vj) {
    int row = rowBase + vj + half * 8;
    float bias;
    if (row == 0) {
      bias = cw[0] * token[col];            // X row 0 is zero -> pure token
    } else {
      int reg = region_of(row);
      bias = rwTab[reg * 2 + 0] * b_seq[col] + rwTab[reg * 2 + 1] * b_pos[col];
    }
    out[(size_t)row * 256 + col] = acc[vj] + bias;
  }
}

// ---------------------------------------------------------------------------
// z fill: 512 MB streaming store, non-temporal 128-bit stores.
// grid (128, 1024): blockIdx.y = i; each block covers 8 j's x 32 float4's.
// ---------------------------------------------------------------------------
__global__ void z_fill(const float* __restrict__ zvec, float* __restrict__ outz) {
  int i  = blockIdx.y;
  int t  = blockIdx.x * blockDim.x + threadIdx.x; // 0..32767
  int c4 = t & 31;
  int j  = t >> 5;

  int ri = region_of(i), rj = region_of(j);
  int pid;
  if (ri == 0 || rj == 0) {
    pid = ((ri | rj) == 0) ? 0 : 1;                 // collapse interaction
  } else if (ri == 1 && rj == 1) {
    int di = i - j; di = di < 0 ? -di : di;
    pid = (di == 0) ? 0 : ((di == 1) ? 2 : 3);      // hd-hd
  } else if (ri == 1 || rj == 1) {
    pid = 4;                                        // hd-cond
  } else if (ri == rj) {
    pid = 3 + ri;                                   // same cond region
  } else {
    int m1 = (ri < rj ? ri : rj) - 2;
    int m2 = (ri > rj ? ri : rj) - 2;
    pid = 11 + m1 * 6 - (m1 * (m1 + 1)) / 2 + (m2 - m1 - 1); // cross pairs
  }

  v4f v = *(const v4f*)(zvec + pid * 128 + c4 * 4); // hot 16KB table, cache-resident
  v4f* dst = (v4f*)(outz + ((size_t)i * 1024 + j) * 128 + (size_t)c4 * 4);
  __builtin_nontemporal_store(v, dst);              // TH=NT: 512MB stream > L2
}

// ---------------------------------------------------------------------------

extern "C" void kernel_launch(void* const* d_in, const int* in_sizes, int n_in,
                              void* d_out, int out_size, void* d_ws,
                              size_t ws_size, hipStream_t stream) {
  const float* hd      = (const float*)d_in[0];
  const int*   hd_idx  = (const int*)  d_in[1];
  const float* mhc     = (const float*)d_in[2];
  const int*   mhc_idx = (const int*)  d_in[3];
  const float* pep     = (const float*)d_in[4];
  const int*   pep_idx = (const int*)  d_in[5];
  const float* lv      = (const float*)d_in[6];
  const int*   lv_idx  = (const int*)  d_in[7];
  const float* lj      = (const float*)d_in[8];
  const int*   lj_idx  = (const int*)  d_in[9];
  const float* hv      = (const float*)d_in[10];
  const int*   hv_idx  = (const int*)  d_in[11];
  const float* hj      = (const float*)d_in[12];
  const int*   hj_idx  = (const int*)  d_in[13];
  const int*   mask    = (const int*)  d_in[14];
  const float* W_seq   = (const float*)d_in[15];
  const float* b_seq   = (const float*)d_in[16];
  const float* W_pos   = (const float*)d_in[17];
  const float* b_pos   = (const float*)d_in[18];
  const float* W_p1    = (const float*)d_in[19];
  const float* b_p1    = (const float*)d_in[20];
  const float* W_p2    = (const float*)d_in[21];
  const float* b_p2    = (const float*)d_in[22];
  const float* token   = (const float*)d_in[23];
  const float* cw      = (const float*)d_in[24];
  const float* rw_hd   = (const float*)d_in[25];
  const float* rw_mhc  = (const float*)d_in[26];
  const float* rw_pep  = (const float*)d_in[27];
  const float* rw_lv   = (const float*)d_in[28];
  const float* rw_lj   = (const float*)d_in[29];
  const float* rw_hv   = (const float*)d_in[30];
  const float* rw_hj   = (const float*)d_in[31];

  float* ws    = (float*)d_ws;
  float* X     = ws;                      // 1024*96  = 98304 floats
  float* W     = X + 1024 * 96;           // 96*256   = 24576 floats
  float* zvec  = W + 96 * 256;            // 32*128   = 4096 floats
  float* rwTab = zvec + 32 * 128;         // 16 floats

  float* out_s = (float*)d_out;           // 1024*256
  float* out_z = out_s + 1024 * 256;      // 1024*1024*128

  prep_rw<<<1, 16, 0, stream>>>(rw_hd, rw_mhc, rw_pep, rw_lv, rw_lj, rw_hv,
                                rw_hj, rwTab);
  build_w<<<96, 256, 0, stream>>>(W_seq, W_pos, W);
  build_zvec<<<32, 128, 0, stream>>>(W_p1, b_p1, W_p2, b_p2, zvec);
  build_x<<<1024, 96, 0, stream>>>(hd, hd_idx, mhc, mhc_idx, pep, pep_idx,
                                   lv, lv_idx, lj, lj_idx, hv, hv_idx,
                                   hj, hj_idx, mask, rw_hd, rw_mhc, rw_pep,
                                   rw_lv, rw_lj, rw_hv, rw_hj, X);
  gemm_s<<<dim3(16, 64), 32, 0, stream>>>(X, W, b_seq, b_pos, token, cw,
                                          rwTab, out_s);
  z_fill<<<dim3(128, 1024), 256, 0, stream>>>(zvec, out_z);
}